// S4D_50285477102161
// MI455X (gfx1250) — compile-verified
//
#include <hip/hip_runtime.h>

#define T_LEN 4096
#define B_SZ 2
#define H_SZ 2048
#define N2 8
#define LC 16
#define NCHUNK (T_LEN / LC)   // 256 chunks per sequence
#define NCOL NCHUNK           // 256 columns per (h,b) workgroup

typedef __attribute__((ext_vector_type(2))) float v2f;
typedef __attribute__((ext_vector_type(8))) float v8f;

// One workgroup per (h, b). Chunked diagonal-SSM:
//   S_loc = W * X          (WMMA f32 16x16x4, K=16)
//   chunk scan: s <- lam^16 * s + s_loc   (8-lane serial scan in LDS)
//   Y = T * X + U * S_init (WMMA) + D*x residual
__global__ __launch_bounds__(128) void s4d_chunked_wmma(
    const float* __restrict__ x, const float* __restrict__ Dp,
    const float* __restrict__ log_dt, const float* __restrict__ A_real_log,
    const float* __restrict__ A_imag, const float* __restrict__ C_re,
    const float* __restrict__ C_im, float* __restrict__ out)
{
  __shared__ float Xs[LC * NCOL];   // 16 KB: X tile, Xs[j*NCOL + c]
  __shared__ float Ss[LC * NCOL];   // 16 KB: S_loc -> S_init (rows 2n=Re, 2n+1=Im)
  __shared__ float Wm[LC * LC];
  __shared__ float Tm[LC * LC];
  __shared__ float Um[LC * LC];
  __shared__ float kpart[N2 * LC];
  __shared__ float kfull[LC];
  __shared__ float lam16s[2 * N2];

  const int h    = blockIdx.x;
  const int b    = blockIdx.y;
  const int tid  = threadIdx.x;
  const int wave = tid >> 5;
  const int lane = tid & 31;
  const int mrow = lane & 15;       // M (or N) index within fragment
  const int hi   = lane >> 4;       // lane-half selects K (A/B) or M+8 (C/D)

  // ---- phase 0a: per-mode constants (8 lanes, one complex mode each) ----
  if (tid < N2) {
    const int n = tid;
    float dt = expf(log_dt[h]);
    float Ar = -expf(A_real_log[h * N2 + n]);
    float Ai = A_imag[h * N2 + n];
    float dr = Ar * dt, di = Ai * dt;
    float e  = expf(dr);
    float lr = e * cosf(di), li = e * sinf(di);          // lambda = exp(dt*A)
    float nr = lr - 1.0f, ni = li;                       // exp(dtA) - 1
    float inv = 1.0f / (Ar * Ar + Ai * Ai);
    float ir = Ar * inv, ii = -Ai * inv;                 // 1/A
    float fr = nr * ir - ni * ii;
    float fi = nr * ii + ni * ir;
    float cr = C_re[h * N2 + n], ci = C_im[h * N2 + n];
    float Cdr = cr * fr - ci * fi;                       // discretized C
    float Cdi = cr * fi + ci * fr;

    float pr[LC + 1], pim[LC + 1];                       // lambda^p, p=0..16
    pr[0] = 1.f; pim[0] = 0.f;
#pragma unroll
    for (int p = 1; p <= LC; p++) {
      pr[p]  = pr[p-1] * lr - pim[p-1] * li;
      pim[p] = pr[p-1] * li + pim[p-1] * lr;
    }
#pragma unroll
    for (int i = 0; i < LC; i++) {                       // W: local end-state
      Wm[(2*n  )*LC + i] = pr [LC-1-i];
      Wm[(2*n+1)*LC + i] = pim[LC-1-i];
    }
#pragma unroll
    for (int j = 0; j < LC; j++) {                       // U: state injection
      Um[j*LC + 2*n  ] =  2.f * (Cdr * pr [j+1] - Cdi * pim[j+1]);
      Um[j*LC + 2*n+1] = -2.f * (Cdr * pim[j+1] + Cdi * pr [j+1]);
    }
#pragma unroll
    for (int d = 0; d < LC; d++)                         // conv kernel partials
      kpart[n*LC + d] = 2.f * (Cdr * pr[d] - Cdi * pim[d]);
    lam16s[2*n] = pr[LC]; lam16s[2*n+1] = pim[LC];       // lambda^16
  }

  // ---- phase 1: gather X tile (strided; whole x lives in 192MB L2) ----
  for (int idx = tid; idx < LC * NCOL; idx += 128) {
    int j = idx >> 8;                 // time-in-chunk
    int c = idx & (NCOL - 1);         // chunk
    int t = c * LC + j;
    Xs[idx] = x[((size_t)t * B_SZ + b) * H_SZ + h];
  }
  __syncthreads();

  if (tid < LC) {                     // reduce kernel partials over modes
    float s = 0.f;
#pragma unroll
    for (int n = 0; n < N2; n++) s += kpart[n*LC + tid];
    kfull[tid] = s;
  }
  __syncthreads();
  for (int idx = tid; idx < LC*LC; idx += 128) {         // Toeplitz matrix
    int j = idx >> 4, i = idx & 15;
    Tm[idx] = (i <= j) ? kfull[j - i] : 0.f;
  }
  __syncthreads();

  // ---- phase 2: S_loc = W * X  (each wave: 4 column tiles of 16 chunks) ----
  for (int it = 0; it < NCOL / 64; it++) {
    int col0 = (wave * 4 + it) * 16;
    v8f acc = {};
#pragma unroll
    for (int kk = 0; kk < 4; kk++) {
      int k0 = kk*4 + 2*hi;
      v2f a, bf;
      a.x  = Wm[mrow*LC + k0];        a.y  = Wm[mrow*LC + k0 + 1];
      bf.x = Xs[k0*NCOL + col0 + mrow];
      bf.y = Xs[(k0+1)*NCOL + col0 + mrow];
      acc = __builtin_amdgcn_wmma_f32_16x16x4_f32(false, a, false, bf,
                                                  (short)0, acc, false, false);
    }
#pragma unroll
    for (int r = 0; r < 8; r++)
      Ss[(r + 8*hi)*NCOL + col0 + mrow] = acc[r];
  }
  __syncthreads();

  // ---- phase 3: chunk-level scan S_loc -> S_init (8 lanes, serial 256) ----
  if (tid < N2) {
    int n = tid;
    float lr = lam16s[2*n], li = lam16s[2*n+1];
    float sr = 0.f, si = 0.f;
    for (int c = 0; c < NCHUNK; c++) {
      float tr = Ss[(2*n  )*NCOL + c];
      float ti = Ss[(2*n+1)*NCOL + c];
      Ss[(2*n  )*NCOL + c] = sr;      // state entering chunk c
      Ss[(2*n+1)*NCOL + c] = si;
      float nsr = lr*sr - li*si + tr; // s <- lam^16 * s + s_loc
      si        = lr*si + li*sr + ti;
      sr        = nsr;
    }
  }
  __syncthreads();

  // ---- phase 4: Y = T*X + U*S_init (+ residual), scatter-store ----
  const float Dh = Dp[h];
  for (int it = 0; it < NCOL / 64; it++) {
    int col0 = (wave * 4 + it) * 16;
    v8f acc = {};
#pragma unroll
    for (int kk = 0; kk < 4; kk++) {  // intra-chunk causal conv
      int k0 = kk*4 + 2*hi;
      v2f a, bf;
      a.x  = Tm[mrow*LC + k0];        a.y  = Tm[mrow*LC + k0 + 1];
      bf.x = Xs[k0*NCOL + col0 + mrow];
      bf.y = Xs[(k0+1)*NCOL + col0 + mrow];
      acc = __builtin_amdgcn_wmma_f32_16x16x4_f32(false, a, false, bf,
                                                  (short)0, acc, false, false);
    }
#pragma unroll
    for (int kk = 0; kk < 4; kk++) {  // carried-state contribution
      int k0 = kk*4 + 2*hi;
      v2f a, bf;
      a.x  = Um[mrow*LC + k0];        a.y  = Um[mrow*LC + k0 + 1];
      bf.x = Ss[k0*NCOL + col0 + mrow];
      bf.y = Ss[(k0+1)*NCOL + col0 + mrow];
      acc = __builtin_amdgcn_wmma_f32_16x16x4_f32(false, a, false, bf,
                                                  (short)0, acc, false, false);
    }
    int c = col0 + mrow;
#pragma unroll
    for (int r = 0; r < 8; r++) {
      int j = r + 8*hi;
      int t = c * LC + j;
      out[((size_t)t * B_SZ + b) * H_SZ + h] = acc[r] + Dh * Xs[j*NCOL + c];
    }
  }
}

extern "C" void kernel_launch(void* const* d_in, const int* in_sizes, int n_in,
                              void* d_out, int out_size, void* d_ws, size_t ws_size,
                              hipStream_t stream) {
  const float* x       = (const float*)d_in[0];
  const float* Dp      = (const float*)d_in[1];
  const float* log_dt  = (const float*)d_in[2];
  const float* Arl     = (const float*)d_in[3];
  const float* Aim     = (const float*)d_in[4];
  const float* Cre     = (const float*)d_in[5];
  const float* Cim     = (const float*)d_in[6];
  float* out           = (float*)d_out;

  dim3 grid(H_SZ, B_SZ);   // 4096 workgroups
  dim3 block(128);         // 4 wave32s
  s4d_chunked_wmma<<<grid, block, 0, stream>>>(x, Dp, log_dt, Arl, Aim, Cre, Cim, out);
}